// AGCL_28020366639450
// MI455X (gfx1250) — compile-verified
//
#include <hip/hip_runtime.h>
#include <hip/hip_bf16.h>
#include <stdint.h>

#define N_   4
#define C_   256
#define H_   64
#define W_   256
#define G_   4
#define CG_  64
#define S_   9
#define HW_  (H_ * W_)

// ---------------------------------------------------------------------------
// Transpose fmap2 NCHW -> NHWC using CDNA5 async global->LDS B128 copies
// (tracked on ASYNCcnt). Tile: 32 channels x 128 pixels, block (32,8).
// Row pitch 132 floats: keeps each LDS B128 write 16B-aligned (132*4 = 528,
// multiple of 16) while limiting store-phase reads to 2-way bank conflicts.
// ---------------------------------------------------------------------------
__global__ __launch_bounds__(256) void t2_nhwc(const float* __restrict__ in,
                                               float* __restrict__ out) {
    __shared__ float tile[32][132];
    const int tx = threadIdx.x;        // 0..31
    const int ty = threadIdx.y;        // 0..7
    const int p0 = blockIdx.x * 128;   // pixel tile origin (h*W+w flattened)
    const int c0 = blockIdx.y * 32;    // channel tile origin
    const int n  = blockIdx.z;
    const float* src = in  + (size_t)n * C_ * HW_;
    float*       dst = out + (size_t)n * HW_ * C_;

#pragma unroll
    for (int j = 0; j < 4; ++j) {
        const int cl = ty + j * 8;     // channel within tile
        unsigned long long ga =
            (unsigned long long)(uintptr_t)(src + (size_t)(c0 + cl) * HW_ + p0 + tx * 4);
        unsigned int la = (unsigned int)(uintptr_t)&tile[cl][tx * 4];
        // CDNA5 async copy: 16B global -> LDS, tracked on ASYNCcnt
        asm volatile("global_load_async_to_lds_b128 %0, %1, off"
                     :: "v"(la), "v"(ga)
                     : "memory");
    }
    asm volatile("s_wait_asynccnt 0" ::: "memory");
    __syncthreads();

#pragma unroll
    for (int j = 0; j < 16; ++j) {
        const int pl = ty + j * 8;     // pixel within tile
        dst[(size_t)(p0 + pl) * C_ + (c0 + tx)] = tile[tx][pl];
    }
}

// ---------------------------------------------------------------------------
// Main correlation kernel. One thread = one pixel (w = threadIdx.x),
// block = one image row, grid = (H, group, N).
// CHLAST=true : fmap2 is NHWC (transposed scratch), float4 tap gathers.
// CHLAST=false: fmap2 is original NCHW, scalar tap gathers (ws fallback).
// ---------------------------------------------------------------------------
template <bool CHLAST>
__global__ __launch_bounds__(256) void corr_kernel(
    const float* __restrict__ f1,    // (N,C,H,W)
    const float* __restrict__ f2,    // CHLAST ? (N,H,W,C) : (N,C,H,W)
    const float* __restrict__ flow,  // (N,2,H,W)
    const float* __restrict__ eo,    // (N,18,H,W)
    float* __restrict__ outp)        // (N,36,H,W)
{
    const int w  = threadIdx.x;
    const int h  = blockIdx.x;
    const int g  = blockIdx.y;
    const int n  = blockIdx.z;
    const int ph = h * W_ + w;

    const float fx = flow[(size_t)(n * 2 + 0) * HW_ + ph];
    const float fy = flow[(size_t)(n * 2 + 1) * HW_ + ph];
    const float cx = (float)w + fx;
    const float cy = (float)h + fy;

    // Precompute per-offset tap addresses and (validity-folded) bilinear weights.
    int   o [S_][4];
    float wt[S_][4];
#pragma unroll
    for (int s = 0; s < S_; ++s) {
        const float ex = eo[(size_t)(n * 18 + 2 * s    ) * HW_ + ph];
        const float ey = eo[(size_t)(n * 18 + 2 * s + 1) * HW_ + ph];
        const float x  = cx + (float)(s - 4) + ex;
        const float y  = cy + ey;
        const float x0f = floorf(x), y0f = floorf(y);
        const float x1f = x0f + 1.0f, y1f = y0f + 1.0f;
        const float wx1 = x - x0f, wx0 = 1.0f - wx1;
        const float wy1 = y - y0f, wy0 = 1.0f - wy1;
        const float vx0 = (x0f >= 0.0f && x0f <= (float)(W_ - 1)) ? 1.0f : 0.0f;
        const float vx1 = (x1f >= 0.0f && x1f <= (float)(W_ - 1)) ? 1.0f : 0.0f;
        const float vy0 = (y0f >= 0.0f && y0f <= (float)(H_ - 1)) ? 1.0f : 0.0f;
        const float vy1 = (y1f >= 0.0f && y1f <= (float)(H_ - 1)) ? 1.0f : 0.0f;
        const int ix0 = (int)fminf(fmaxf(x0f, 0.0f), (float)(W_ - 1));
        const int ix1 = (int)fminf(fmaxf(x1f, 0.0f), (float)(W_ - 1));
        const int iy0 = (int)fminf(fmaxf(y0f, 0.0f), (float)(H_ - 1));
        const int iy1 = (int)fminf(fmaxf(y1f, 0.0f), (float)(H_ - 1));
        if (CHLAST) {
            o[s][0] = (iy0 * W_ + ix0) * C_;
            o[s][1] = (iy0 * W_ + ix1) * C_;
            o[s][2] = (iy1 * W_ + ix0) * C_;
            o[s][3] = (iy1 * W_ + ix1) * C_;
        } else {
            o[s][0] = iy0 * W_ + ix0;
            o[s][1] = iy0 * W_ + ix1;
            o[s][2] = iy1 * W_ + ix0;
            o[s][3] = iy1 * W_ + ix1;
        }
        wt[s][0] = wx0 * wy0 * vx0 * vy0;
        wt[s][1] = wx1 * wy0 * vx1 * vy0;
        wt[s][2] = wx0 * wy1 * vx0 * vy1;
        wt[s][3] = wx1 * wy1 * vx1 * vy1;
    }

    const float* f1n = f1 + ((size_t)n * C_ + (size_t)g * CG_) * HW_ + ph; // +c*HW_
    const float* f2n = CHLAST ? (f2 + (size_t)n * HW_ * C_)
                              : (f2 + (size_t)n * C_  * HW_);

    float acc[S_];
#pragma unroll
    for (int s = 0; s < S_; ++s) acc[s] = 0.0f;

    for (int cc = 0; cc < CG_; cc += 4) {
        // fmap1: NCHW, consecutive lanes -> consecutive w -> fully coalesced
        const float a0 = f1n[(size_t)(cc + 0) * HW_];
        const float a1 = f1n[(size_t)(cc + 1) * HW_];
        const float a2 = f1n[(size_t)(cc + 2) * HW_];
        const float a3 = f1n[(size_t)(cc + 3) * HW_];
        const int cb = g * CG_ + cc;

#pragma unroll
        for (int s = 0; s < S_; ++s) {
            float r0, r1, r2, r3;
            if (CHLAST) {
                const float4 t0 = *reinterpret_cast<const float4*>(f2n + o[s][0] + cb);
                const float4 t1 = *reinterpret_cast<const float4*>(f2n + o[s][1] + cb);
                const float4 t2 = *reinterpret_cast<const float4*>(f2n + o[s][2] + cb);
                const float4 t3 = *reinterpret_cast<const float4*>(f2n + o[s][3] + cb);
                r0 = t0.x * wt[s][0]; r0 = fmaf(t1.x, wt[s][1], r0);
                r0 = fmaf(t2.x, wt[s][2], r0); r0 = fmaf(t3.x, wt[s][3], r0);
                r1 = t0.y * wt[s][0]; r1 = fmaf(t1.y, wt[s][1], r1);
                r1 = fmaf(t2.y, wt[s][2], r1); r1 = fmaf(t3.y, wt[s][3], r1);
                r2 = t0.z * wt[s][0]; r2 = fmaf(t1.z, wt[s][1], r2);
                r2 = fmaf(t2.z, wt[s][2], r2); r2 = fmaf(t3.z, wt[s][3], r2);
                r3 = t0.w * wt[s][0]; r3 = fmaf(t1.w, wt[s][1], r3);
                r3 = fmaf(t2.w, wt[s][2], r3); r3 = fmaf(t3.w, wt[s][3], r3);
            } else {
                const float* b0 = f2n + (size_t)(cb + 0) * HW_;
                const float* b1 = f2n + (size_t)(cb + 1) * HW_;
                const float* b2 = f2n + (size_t)(cb + 2) * HW_;
                const float* b3 = f2n + (size_t)(cb + 3) * HW_;
                r0 = b0[o[s][0]] * wt[s][0]; r0 = fmaf(b0[o[s][1]], wt[s][1], r0);
                r0 = fmaf(b0[o[s][2]], wt[s][2], r0); r0 = fmaf(b0[o[s][3]], wt[s][3], r0);
                r1 = b1[o[s][0]] * wt[s][0]; r1 = fmaf(b1[o[s][1]], wt[s][1], r1);
                r1 = fmaf(b1[o[s][2]], wt[s][2], r1); r1 = fmaf(b1[o[s][3]], wt[s][3], r1);
                r2 = b2[o[s][0]] * wt[s][0]; r2 = fmaf(b2[o[s][1]], wt[s][1], r2);
                r2 = fmaf(b2[o[s][2]], wt[s][2], r2); r2 = fmaf(b2[o[s][3]], wt[s][3], r2);
                r3 = b3[o[s][0]] * wt[s][0]; r3 = fmaf(b3[o[s][1]], wt[s][1], r3);
                r3 = fmaf(b3[o[s][2]], wt[s][2], r3); r3 = fmaf(b3[o[s][3]], wt[s][3], r3);
            }
            acc[s] = fmaf(a0, r0, acc[s]);
            acc[s] = fmaf(a1, r1, acc[s]);
            acc[s] = fmaf(a2, r2, acc[s]);
            acc[s] = fmaf(a3, r3, acc[s]);
        }
    }

    // out[n, g*9+s, h, w]
    float* op = outp + (size_t)(n * G_ * S_ + g * S_) * HW_ + ph;
#pragma unroll
    for (int s = 0; s < S_; ++s) op[(size_t)s * HW_] = acc[s] * (1.0f / (float)CG_);
}

extern "C" void kernel_launch(void* const* d_in, const int* in_sizes, int n_in,
                              void* d_out, int out_size, void* d_ws, size_t ws_size,
                              hipStream_t stream) {
    const float* f1   = (const float*)d_in[0];
    const float* f2   = (const float*)d_in[1];
    const float* flow = (const float*)d_in[2];
    const float* eo   = (const float*)d_in[3];
    float* out = (float*)d_out;

    const size_t need = (size_t)N_ * HW_ * C_ * sizeof(float); // 64 MB NHWC copy of fmap2

    dim3 blk(256);
    dim3 grd(H_, G_, N_);

    if (d_ws != nullptr && ws_size >= need) {
        float* f2t = (float*)d_ws;
        dim3 tb(32, 8);
        dim3 tg(HW_ / 128, C_ / 32, N_);
        t2_nhwc<<<tg, tb, 0, stream>>>(f2, f2t);
        corr_kernel<true><<<grd, blk, 0, stream>>>(f1, f2t, flow, eo, out);
    } else {
        corr_kernel<false><<<grd, blk, 0, stream>>>(f1, f2, flow, eo, out);
    }
}